// newton_modify_65317862638029
// MI455X (gfx1250) — compile-verified
//
#include <hip/hip_runtime.h>

#define HW    192
#define PLANE (HW * HW)   // 36864 pixels

typedef float v2f __attribute__((ext_vector_type(2)));
typedef float v8f __attribute__((ext_vector_type(8)));

// ---------------------------------------------------------------------------
// Build dense (zero-padded, block-diagonal) Kpad x 16 weight matrix for a
// grouped 3x3 conv with OIHW weights: B[k][co], k = ci*9 + (dy*3+dx).
// ---------------------------------------------------------------------------
__global__ void build_dense_k(const float* __restrict__ w, float* __restrict__ Bw,
                              int Cout, int CinPG, int groups, int CinTot, int Kpad)
{
  int t = blockIdx.x * blockDim.x + threadIdx.x;
  int total = Kpad * 16;
  if (t >= total) return;
  int k = t >> 4, co = t & 15;
  float v = 0.0f;
  if (k < CinTot * 9 && co < Cout) {
    int ci = k / 9, tt = k - ci * 9;
    int g = ci / CinPG;
    int CoutPG = Cout / groups;
    if (co / CoutPG == g) {
      int ciLoc = ci - g * CinPG;
      v = w[(co * CinPG + ciLoc) * 9 + tt];
    }
  }
  Bw[t] = v;
}

// ---------------------------------------------------------------------------
// Grouped circular 3x3 conv as dense im2col GEMM using V_WMMA_F32_16X16X4_F32.
// Compile-time KACT/COUT: fully unrolled K loop (one v_wmma per 4 K), all
// ci/dy/dx address math constant-folded, tail masking folded to cndmask.
//   A (16x4 f32): lane L holds K = 2*(L>=16)+{0,1} for pixel (L&15)
//   B (4x16 f32): VGPR j, lanes 0-15 -> row k0+j, lanes 16-31 -> row k0+j+2
//   C/D (16x16 f32): VGPR v, lane L -> pixel v+8*(L>=16), channel (L&15)
// ---------------------------------------------------------------------------
template <int KACT, int COUT>
__global__ __launch_bounds__(256) void conv_wmma_t(
    const float* __restrict__ in, const float* __restrict__ Bw,
    float* __restrict__ out)
{
  constexpr int KPAD = (KACT + 3) & ~3;
  const int wave = threadIdx.x >> 5;
  const int lane = threadIdx.x & 31;
  const int tile = blockIdx.x * 8 + wave;   // 2304 tiles total
  const int base = tile * 16;
  const int half = lane >> 4;
  const int n    = lane & 15;

  // pixel whose A-row this lane supplies
  const int p = base + n;
  const int h = p / HW, w = p - h * HW;
  const int hm = (h == 0)      ? (HW - 1) : h - 1;
  const int hp = (h == HW - 1) ? 0        : h + 1;
  const int wm = (w == 0)      ? (HW - 1) : w - 1;
  const int wp = (w == HW - 1) ? 0        : w + 1;
  int rofs[3] = { hm * HW, h * HW, hp * HW };
  int cofs[3] = { wm, w, wp };

  // compile-time k -> runtime byte-free element offset (constant-folds)
  auto off = [&](int k) -> int {
    int ci = k / 9, t = k - ci * 9;
    return ci * PLANE + rofs[t / 3] + cofs[t % 3];
  };

  const float* Bp = Bw + half * 32 + n;   // row (k0 + 2*half), col n

  v8f acc = {};
#pragma unroll
  for (int k0 = 0; k0 < KPAD; k0 += 4) {
    v2f a, b;
    {
      constexpr bool dummy = true; (void)dummy;
      const bool s0v = (k0 + 0) < KACT, s2v = (k0 + 2) < KACT;
      int   oa = half ? (s2v ? off(k0 + 2) : 0) : (s0v ? off(k0 + 0) : 0);
      float sa = half ? (s2v ? 1.0f : 0.0f)     : (s0v ? 1.0f : 0.0f);
      a.x = in[oa] * sa;
    }
    {
      const bool s1v = (k0 + 1) < KACT, s3v = (k0 + 3) < KACT;
      int   ob = half ? (s3v ? off(k0 + 3) : 0) : (s1v ? off(k0 + 1) : 0);
      float sb = half ? (s3v ? 1.0f : 0.0f)     : (s1v ? 1.0f : 0.0f);
      a.y = in[ob] * sb;
    }
    b.x = Bp[k0 * 16];        // row k0 + 2*half
    b.y = Bp[k0 * 16 + 16];   // row k0 + 2*half + 1
    // D = A x B + C   (emits v_wmma_f32_16x16x4_f32)
    acc = __builtin_amdgcn_wmma_f32_16x16x4_f32(false, a, false, b,
                                                (short)0, acc, false, false);
  }

  if (n < COUT) {
    float* op = out + (size_t)n * PLANE + base + half * 8;
#pragma unroll
    for (int v = 0; v < 8; ++v) op[v] = acc[v];
  }
}

// ---------------------------------------------------------------------------
// PWL: t = lam/sigma + fu ; th/dri = sum_k sin(t*pi*(k+1))*wa + cos(..)*wb + b
// Series weights staged through LDS.
// ---------------------------------------------------------------------------
__global__ __launch_bounds__(256) void pwl_k(
    const float* __restrict__ lam, const float* __restrict__ fu,
    const float* __restrict__ sig,
    const float* __restrict__ wa, const float* __restrict__ wb,
    const float* __restrict__ bias,
    float* __restrict__ th, float* __restrict__ dri)
{
  __shared__ float swa[600], swb[600], sb[24];
  for (int t = threadIdx.x; t < 600; t += 256) { swa[t] = wa[t]; swb[t] = wb[t]; }
  if (threadIdx.x < 24) sb[threadIdx.x] = bias[threadIdx.x];
  __syncthreads();
  int idx = blockIdx.x * 256 + threadIdx.x;
  if (idx >= 12 * PLANE) return;
  int c = idx / PLANE;
  float xv = lam[idx] / sig[c] + fu[idx];
  float a0 = sb[c], a1 = sb[12 + c];
  const float* wa0 = swa + c * 25;       const float* wb0 = swb + c * 25;
  const float* wa1 = swa + 300 + c * 25; const float* wb1 = swb + 300 + c * 25;
#pragma unroll 5
  for (int k = 0; k < 25; ++k) {
    float s, co;
    __sincosf(xv * (3.14159265358979323846f * (float)(k + 1)), &s, &co);
    a0 = __builtin_fmaf(s, wa0[k], __builtin_fmaf(co, wb0[k], a0));
    a1 = __builtin_fmaf(s, wa1[k], __builtin_fmaf(co, wb1[k], a1));
  }
  th[idx] = a0;
  if (dri) dri[idx] = a1;
}

// ---------------------------------------------------------------------------
// Elementwise helpers
// ---------------------------------------------------------------------------
__global__ void ew_copy_k(float* __restrict__ d, const float* __restrict__ s, int n)
{
  int i = blockIdx.x * blockDim.x + threadIdx.x;
  if (i < n) d[i] = s[i];
}

__global__ void ew_subip_k(float* __restrict__ a, const float* __restrict__ b, int n)
{ // a -= b
  int i = blockIdx.x * blockDim.x + threadIdx.x;
  if (i < n) a[i] = a[i] - b[i];
}

__global__ void ew_onemdri_k(float* __restrict__ a, const float* __restrict__ dri, int n)
{ // a *= (1 - dri)
  int i = blockIdx.x * blockDim.x + threadIdx.x;
  if (i < n) a[i] = a[i] * (1.0f - dri[i]);
}

__global__ void combine_k(const float* __restrict__ m1, const float* __restrict__ m2,
                          const float* __restrict__ sig, const float* __restrict__ u,
                          const float* __restrict__ z, float* __restrict__ out3, int n)
{ // out3 = rgb_sum(m1 + sigma*m2) + u - z   (3 planes)
  int idx = blockIdx.x * blockDim.x + threadIdx.x;
  if (idx >= n) return;
  int r = idx / PLANE, p = idx - r * PLANE;
  float acc = 0.0f;
#pragma unroll
  for (int j = 0; j < 4; ++j) {
    int ch = r * 4 + j;
    acc += m1[ch * PLANE + p] + sig[ch] * m2[ch * PLANE + p];
  }
  out3[idx] = acc + u[idx] - z[idx];
}

__global__ void update_k(float* __restrict__ u, const float* __restrict__ ffo,
                         const float* __restrict__ sig, const float* __restrict__ out3,
                         const float* __restrict__ eta, int ei, int n)
{ // u -= eta * (rgb_sum(sigma*ffo) + out3)
  int idx = blockIdx.x * blockDim.x + threadIdx.x;
  if (idx >= n) return;
  int r = idx / PLANE, p = idx - r * PLANE;
  float acc = 0.0f;
#pragma unroll
  for (int j = 0; j < 4; ++j) {
    int ch = r * 4 + j;
    acc += sig[ch] * ffo[ch * PLANE + p];
  }
  u[idx] = u[idx] - eta[ei] * (acc + out3[idx]);
}

__global__ void lamupd_k(float* __restrict__ lam, const float* __restrict__ s2,
                         const float* __restrict__ pout, const float* __restrict__ sig, int n)
{ // lam += sigma * (s2 - pout)
  int idx = blockIdx.x * blockDim.x + threadIdx.x;
  if (idx >= n) return;
  int c = idx / PLANE;
  lam[idx] = lam[idx] + sig[c] * (s2[idx] - pout[idx]);
}

__global__ void sigma_init_k(float* sig)
{
  if (threadIdx.x < 12) sig[threadIdx.x] = 1.0f;
}

__global__ void sigma_mul_k(float* sig, const float* c)
{
  if (threadIdx.x < 12) sig[threadIdx.x] *= c[threadIdx.x];
}

__global__ void clip_k(float* __restrict__ out, const float* __restrict__ u, int n)
{
  int i = blockIdx.x * blockDim.x + threadIdx.x;
  if (i < n) out[i] = fminf(fmaxf(u[i], 0.0f), 1.0f);
}

// ---------------------------------------------------------------------------
// Host-side helpers
// ---------------------------------------------------------------------------
static inline int ceilDiv(int a, int b) { return (a + b - 1) / b; }

static void simc_run(const float* in3, const float* d1, const float* d2, const float* d3,
                     float* catS, float* out12, hipStream_t s)
{
  // catS planes: [0..2]=input, [3..8]=x1, [9..17]=x2
  ew_copy_k<<<ceilDiv(3 * PLANE, 256), 256, 0, s>>>(catS, in3, 3 * PLANE);
  conv_wmma_t<27, 6><<<288, 256, 0, s>>>(catS, d1, catS + 3 * PLANE);
  conv_wmma_t<81, 9><<<288, 256, 0, s>>>(catS, d2, catS + 9 * PLANE);
  conv_wmma_t<162, 12><<<288, 256, 0, s>>>(catS, d3, out12);
}

static void mimc_run(const float* in12, const float* d1, const float* d2, const float* d3,
                     float* catM, float* out12, hipStream_t s)
{
  // catM planes: [0..11]=input, [12..15]=x1, [16..23]=x2
  ew_copy_k<<<ceilDiv(12 * PLANE, 256), 256, 0, s>>>(catM, in12, 12 * PLANE);
  conv_wmma_t<108, 4><<<288, 256, 0, s>>>(catM, d1, catM + 12 * PLANE);
  conv_wmma_t<144, 8><<<288, 256, 0, s>>>(catM, d2, catM + 16 * PLANE);
  conv_wmma_t<216, 12><<<288, 256, 0, s>>>(catM, d3, out12);
}

extern "C" void kernel_launch(void* const* d_in, const int* in_sizes, int n_in,
                              void* d_out, int out_size, void* d_ws, size_t ws_size,
                              hipStream_t stream)
{
  (void)in_sizes; (void)n_in; (void)out_size; (void)ws_size;

  const float* x     = (const float*)d_in[0];   // (1,3,192,192)
  const float* wA    = (const float*)d_in[1];   // (3,2,12,25)
  const float* wB    = (const float*)d_in[2];   // (3,2,12,25)
  const float* bias  = (const float*)d_in[3];   // (3,2,12)
  const float* cvec  = (const float*)d_in[4];   // (3,12)
  const float* eta   = (const float*)d_in[5];   // (3,3)
  const float* wInit = (const float*)d_in[6];   // (12,1,3,3)
  const float* sw1   = (const float*)d_in[7];   // (3,5,6,3,3,3)
  const float* sw2   = (const float*)d_in[8];   // (3,5,9,3,3,3)
  const float* sw3   = (const float*)d_in[9];   // (3,5,12,3,3,3)
  const float* mw1   = (const float*)d_in[10];  // (3,4,4,3,3,3)
  const float* mw2   = (const float*)d_in[11];  // (3,4,8,2,3,3)
  const float* mw3   = (const float*)d_in[12];  // (3,4,12,2,3,3)
  float* out = (float*)d_out;                   // (1,3,192,192)
  float* W = (float*)d_ws;

  // ---- workspace layout (floats) ----
  size_t off = 0;
  float* dwInit = W + off; off += 28 * 16;
  float* dwS1   = W + off; off += (size_t)15 * 28 * 16;
  float* dwS2   = W + off; off += (size_t)15 * 84 * 16;
  float* dwS3   = W + off; off += (size_t)15 * 164 * 16;
  float* dwM1   = W + off; off += (size_t)12 * 108 * 16;
  float* dwM2   = W + off; off += (size_t)12 * 144 * 16;
  float* dwM3   = W + off; off += (size_t)12 * 216 * 16;
  float* U    = W + off; off += (size_t)3 * PLANE;
  float* LAM  = W + off; off += (size_t)12 * PLANE;
  float* CATS = W + off; off += (size_t)18 * PLANE;
  float* FU   = W + off; off += (size_t)12 * PLANE;
  float* TH   = W + off; off += (size_t)12 * PLANE;
  float* DRI  = W + off; off += (size_t)12 * PLANE;
  float* CATM = W + off; off += (size_t)24 * PLANE;
  float* M1B  = W + off; off += (size_t)12 * PLANE;
  float* M2B  = W + off; off += (size_t)12 * PLANE;
  float* OUT3 = W + off; off += (size_t)3 * PLANE;
  float* FFOS = W + off; off += (size_t)12 * PLANE;
  float* SIG  = W + off; off += 16;

  // ---- build all dense weight matrices ----
  build_dense_k<<<ceilDiv(28 * 16, 256), 256, 0, stream>>>(wInit, dwInit, 12, 1, 3, 3, 28);
  for (int t = 0; t < 15; ++t) {  // simc instances: (i,e) = i*5+e
    build_dense_k<<<ceilDiv(28 * 16, 256),  256, 0, stream>>>(sw1 + t * 162, dwS1 + (size_t)t * 28 * 16,  6, 3, 1,  3,  28);
    build_dense_k<<<ceilDiv(84 * 16, 256),  256, 0, stream>>>(sw2 + t * 243, dwS2 + (size_t)t * 84 * 16,  9, 3, 3,  9,  84);
    build_dense_k<<<ceilDiv(164 * 16, 256), 256, 0, stream>>>(sw3 + t * 324, dwS3 + (size_t)t * 164 * 16, 12, 3, 6, 18, 164);
  }
  for (int t = 0; t < 12; ++t) {  // mimc instances: (i,e) = i*4+e
    build_dense_k<<<ceilDiv(108 * 16, 256), 256, 0, stream>>>(mw1 + t * 108, dwM1 + (size_t)t * 108 * 16,  4, 3,  4, 12, 108);
    build_dense_k<<<ceilDiv(144 * 16, 256), 256, 0, stream>>>(mw2 + t * 144, dwM2 + (size_t)t * 144 * 16,  8, 2,  8, 16, 144);
    build_dense_k<<<ceilDiv(216 * 16, 256), 256, 0, stream>>>(mw3 + t * 216, dwM3 + (size_t)t * 216 * 16, 12, 2, 12, 24, 216);
  }

  // ---- init ----
  ew_copy_k<<<ceilDiv(3 * PLANE, 256), 256, 0, stream>>>(U, x, 3 * PLANE);
  sigma_init_k<<<1, 32, 0, stream>>>(SIG);
  conv_wmma_t<27, 12><<<288, 256, 0, stream>>>(x, dwInit, LAM);

  const int nEl12 = 12 * PLANE, nEl3 = 3 * PLANE;
  const int g12 = ceilDiv(nEl12, 256), g3 = ceilDiv(nEl3, 256);

  for (int i = 0; i < 3; ++i) {
    const float* wa_i = wA + i * 600;
    const float* wb_i = wB + i * 600;
    const float* b_i  = bias + i * 24;
#define SD(e) dwS1 + (size_t)(i * 5 + (e)) * 28 * 16,  dwS2 + (size_t)(i * 5 + (e)) * 84 * 16,  dwS3 + (size_t)(i * 5 + (e)) * 164 * 16
#define MD(e) dwM1 + (size_t)(i * 4 + (e)) * 108 * 16, dwM2 + (size_t)(i * 4 + (e)) * 144 * 16, dwM3 + (size_t)(i * 4 + (e)) * 216 * 16
    for (int e = 0; e < 3; ++e) {
      // f_u = simc(u, sw[-1])
      simc_run(U, SD(4), CATS, FU, stream);
      // th, dri = pwl(lam/sigma + f_u)
      pwl_k<<<g12, 256, 0, stream>>>(LAM, FU, SIG, wa_i, wb_i, b_i, TH, DRI);
      // m1 = mimc(lam, mw[-1])
      mimc_run(LAM, MD(3), CATM, M1B, stream);
      // FU <- f_u - th ; m2 = mimc(FU, mw[-1])
      ew_subip_k<<<g12, 256, 0, stream>>>(FU, TH, nEl12);
      mimc_run(FU, MD(3), CATM, M2B, stream);
      // out3 = rgb_sum(m1 + sigma*m2) + u - z
      combine_k<<<g3, 256, 0, stream>>>(M1B, M2B, SIG, U, x, OUT3, nEl3);
      // ffo = mimc(simc(out3, sw[e]) * (1-dri), mw[e])   (reuse M1B for result)
      simc_run(OUT3, SD(e), CATS, FFOS, stream);
      ew_onemdri_k<<<g12, 256, 0, stream>>>(FFOS, DRI, nEl12);
      mimc_run(FFOS, MD(e), CATM, M1B, stream);
      // u -= eta[i,e] * (rgb_sum(sigma*ffo) + out3)
      update_k<<<g3, 256, 0, stream>>>(U, M1B, SIG, OUT3, eta, i * 3 + e, nEl3);
    }
    // f_o = simc(u, sw[-1]); p_out = pwl series 0 of (lam/sigma + f_o) -> TH
    simc_run(U, SD(4), CATS, FU, stream);
    pwl_k<<<g12, 256, 0, stream>>>(LAM, FU, SIG, wa_i, wb_i, b_i, TH, nullptr);
    // s2 = simc(u, sw[-2]) -> M2B ; lam += sigma*(s2 - p_out)
    simc_run(U, SD(3), CATS, M2B, stream);
    lamupd_k<<<g12, 256, 0, stream>>>(LAM, M2B, TH, SIG, nEl12);
    // sigma *= c[i]
    sigma_mul_k<<<1, 32, 0, stream>>>(SIG, cvec + i * 12);
#undef SD
#undef MD
  }

  clip_k<<<g3, 256, 0, stream>>>(out, U, nEl3);
}